// EnhancedGNNModel_47115791237144
// MI455X (gfx1250) — compile-verified
//
#include <hip/hip_runtime.h>
#include <hip/hip_bf16.h>

typedef __attribute__((ext_vector_type(2))) float v2f;
typedef __attribute__((ext_vector_type(8))) float v8f;

// ---------------------------------------------------------------------------
// Utility: zero-fill
// ---------------------------------------------------------------------------
__global__ void fill_zero_f32(float* __restrict__ p, size_t n) {
    size_t i = blockIdx.x * (size_t)blockDim.x + threadIdx.x;
    size_t stride = (size_t)gridDim.x * blockDim.x;
    for (; i < n; i += stride) p[i] = 0.0f;
}

// ---------------------------------------------------------------------------
// Degree: deg[dst] += 1 over edges, then rdeg = 1/max(deg,1)
// ---------------------------------------------------------------------------
__global__ void deg_count(const int* __restrict__ dst, int ne, float* __restrict__ deg) {
    int e = blockIdx.x * blockDim.x + threadIdx.x;
    if (e < ne) atomicAdd(&deg[dst[e]], 1.0f);
}

__global__ void rdeg_kernel(const float* __restrict__ deg, float* __restrict__ rdeg, int n) {
    int i = blockIdx.x * blockDim.x + threadIdx.x;
    if (i < n) rdeg[i] = 1.0f / fmaxf(deg[i], 1.0f);
}

// ---------------------------------------------------------------------------
// Scatter: msg[dst, c] += x[src, c].  One thread per (edge, channel);
// 32 consecutive lanes cover 32 consecutive channels of one edge, so the
// gather of x[src] is a coalesced 128B burst and atomics hit consecutive
// addresses of one msg row (resolved at L2 — 192MB easily holds msg).
// ch is a power of two (64 or 128).
// ---------------------------------------------------------------------------
__global__ void scatter_add(const float* __restrict__ x,
                            const int* __restrict__ src,
                            const int* __restrict__ dst,
                            int ne, int ch_log2,
                            float* __restrict__ msg) {
    size_t tid = blockIdx.x * (size_t)blockDim.x + threadIdx.x;
    size_t total = (size_t)ne << ch_log2;
    if (tid >= total) return;
    int ch = 1 << ch_log2;
    int e = (int)(tid >> ch_log2);
    int c = (int)(tid & (ch - 1));
    int s = src[e];
    int d = dst[e];
    atomicAdd(&msg[(size_t)d * ch + c], x[(size_t)s * ch + c]);
}

// ---------------------------------------------------------------------------
// Pack the three head weight matrices [128x21],[128x2],[128x5] (lin_l then
// lin_r) into a single [256 x 32] block (cols 28..31 zero) + packed bias.
// ---------------------------------------------------------------------------
__global__ void pack_heads(const float* __restrict__ Wl_a, const float* __restrict__ Wr_a,
                           const float* __restrict__ b_a,
                           const float* __restrict__ Wl_s, const float* __restrict__ Wr_s,
                           const float* __restrict__ b_s,
                           const float* __restrict__ Wl_e, const float* __restrict__ Wr_e,
                           const float* __restrict__ b_e,
                           float* __restrict__ Wpack, float* __restrict__ bpack) {
    int idx = blockIdx.x * blockDim.x + threadIdx.x;
    if (idx >= 256 * 32) return;
    int k = idx >> 5;       // 0..255
    int c = idx & 31;       // 0..31
    int kk = k & 127;
    bool right = (k >= 128);
    float v = 0.0f;
    if (c < 21)       v = (right ? Wr_a : Wl_a)[kk * 21 + c];
    else if (c < 23)  v = (right ? Wr_s : Wl_s)[kk * 2 + (c - 21)];
    else if (c < 28)  v = (right ? Wr_e : Wl_e)[kk * 5 + (c - 23)];
    Wpack[idx] = v;
    if (k == 0) {
        float bv = 0.0f;
        if (c < 21)      bv = b_a[c];
        else if (c < 23) bv = b_s[c - 21];
        else if (c < 28) bv = b_e[c - 23];
        bpack[c] = bv;
    }
}

// ---------------------------------------------------------------------------
// Fused SAGE combine via WMMA:  out = act( mean @ Wl + x @ Wr + b )
// mean[row, k] = msg[row, k] * rdeg[row] is formed during the A-fragment load.
// One wave (32 lanes) computes one 16x16 output tile using
// V_WMMA_F32_16X16X4_F32, looping K in steps of 4 over both terms.
//
// Fragment layouts (ISA 7.12.2, 32-bit):
//   A 16x4 : lane -> M = lane&15 (both halves); VGPR0 holds K = 0 + 2*(lane>>4),
//            VGPR1 holds K = 1 + 2*(lane>>4)
//   B 4x16 : lane -> N = lane&15; VGPR0 holds K = 0 + 2*(lane>>4),
//            VGPR1 holds K = 1 + 2*(lane>>4)
//   C/D    : VGPR i -> M = i + 8*(lane>>4), N = lane&15
//
// mode 0: hidden layer (ReLU, out[row*ncols+col])
// mode 1: heads (no ReLU, demux cols 0..27 into concatenated age|sex|eth)
// EXEC is all-ones for every WMMA (divergence only in the final head store).
// ---------------------------------------------------------------------------
__global__ __launch_bounds__(32)
void sage_combine_wmma(const float* __restrict__ msg,
                       const float* __restrict__ rdeg,
                       const float* __restrict__ xin, int K,
                       const float* __restrict__ Wl,
                       const float* __restrict__ Wr,
                       const float* __restrict__ bias,
                       float* __restrict__ out,
                       int ncols, int mode, int nnodes) {
    const int ntiles = ncols >> 4;
    const int mt = blockIdx.x / ntiles;
    const int nt = blockIdx.x % ntiles;
    const int lane = threadIdx.x;
    const int l15 = lane & 15;
    const int half = lane >> 4;          // 0 or 1
    const int khalf = half * 2;

    const int mrow = mt * 16 + l15;      // this lane's fixed A row
    const int col  = nt * 16 + l15;      // this lane's fixed B column

    const float rd = rdeg[mrow];
    const float* __restrict__ msgrow = msg + (size_t)mrow * K;
    const float* __restrict__ xrow   = xin + (size_t)mrow * K;

    v8f acc = {};

    // term 1: mean @ Wl
    for (int k = 0; k < K; k += 4) {
        const int ka = k + khalf;
        v2f a, b;
        a[0] = msgrow[ka] * rd;
        a[1] = msgrow[ka + 1] * rd;
        b[0] = Wl[(size_t)ka * ncols + col];
        b[1] = Wl[(size_t)(ka + 1) * ncols + col];
        acc = __builtin_amdgcn_wmma_f32_16x16x4_f32(false, a, false, b,
                                                    (short)0, acc, false, false);
    }
    // term 2: x @ Wr
    for (int k = 0; k < K; k += 4) {
        const int ka = k + khalf;
        v2f a, b;
        a[0] = xrow[ka];
        a[1] = xrow[ka + 1];
        b[0] = Wr[(size_t)ka * ncols + col];
        b[1] = Wr[(size_t)(ka + 1) * ncols + col];
        acc = __builtin_amdgcn_wmma_f32_16x16x4_f32(false, a, false, b,
                                                    (short)0, acc, false, false);
    }

    const float bcol = bias[col];
    if (mode == 0) {
        // hidden layer: ReLU, dense [nnodes x ncols] row-major
        #pragma unroll
        for (int i = 0; i < 8; ++i) {
            int r = mt * 16 + i + 8 * half;
            float v = fmaxf(acc[i] + bcol, 0.0f);
            out[(size_t)r * ncols + col] = v;
        }
    } else {
        // heads: demux col into age[ n x21 ] | sex[ n x2 ] | eth[ n x5 ]
        const size_t off_sex = (size_t)nnodes * 21;
        const size_t off_eth = (size_t)nnodes * 23;
        #pragma unroll
        for (int i = 0; i < 8; ++i) {
            int r = mt * 16 + i + 8 * half;
            float v = acc[i] + bcol;
            if (col < 21)       out[(size_t)r * 21 + col] = v;
            else if (col < 23)  out[off_sex + (size_t)r * 2 + (col - 21)] = v;
            else if (col < 28)  out[off_eth + (size_t)r * 5 + (col - 23)] = v;
            // cols 28..31: padding, dropped
        }
    }
}

// ---------------------------------------------------------------------------
// Launch
// ---------------------------------------------------------------------------
extern "C" void kernel_launch(void* const* d_in, const int* in_sizes, int n_in,
                              void* d_out, int out_size, void* d_ws, size_t ws_size,
                              hipStream_t stream) {
    const int IN_CH = 64, HID = 128;
    const int nn = in_sizes[0] / IN_CH;   // 50000
    const int ne = in_sizes[1];           // 800000

    const float* x   = (const float*)d_in[0];
    const int* esrc  = (const int*)d_in[1];
    const int* edst  = (const int*)d_in[2];
    const float* Wl1 = (const float*)d_in[3],  *Wr1 = (const float*)d_in[4],  *b1 = (const float*)d_in[5];
    const float* Wl2 = (const float*)d_in[6],  *Wr2 = (const float*)d_in[7],  *b2 = (const float*)d_in[8];
    const float* Wl3 = (const float*)d_in[9],  *Wr3 = (const float*)d_in[10], *b3 = (const float*)d_in[11];
    const float* WlA = (const float*)d_in[12], *WrA = (const float*)d_in[13], *bA = (const float*)d_in[14];
    const float* WlS = (const float*)d_in[15], *WrS = (const float*)d_in[16], *bS = (const float*)d_in[17];
    const float* WlE = (const float*)d_in[18], *WrE = (const float*)d_in[19], *bE = (const float*)d_in[20];
    float* out = (float*)d_out;

    // workspace carve-up (floats)
    float* deg   = (float*)d_ws;                    // nn
    float* rdeg  = deg + nn;                        // nn
    float* msg   = rdeg + nn;                       // nn*128
    float* hA    = msg + (size_t)nn * HID;          // nn*128
    float* hB    = hA + (size_t)nn * HID;           // nn*128
    float* Wpack = hB + (size_t)nn * HID;           // 256*32
    float* bpack = Wpack + 256 * 32;                // 32

    const int ZB = 2048, ZT = 256;

    // degrees (shared by every layer)
    fill_zero_f32<<<ZB, ZT, 0, stream>>>(deg, (size_t)nn);
    deg_count<<<(ne + 255) / 256, 256, 0, stream>>>(edst, ne, deg);
    rdeg_kernel<<<(nn + 255) / 256, 256, 0, stream>>>(deg, rdeg, nn);

    // packed head weights
    pack_heads<<<(256 * 32 + 255) / 256, 256, 0, stream>>>(
        WlA, WrA, bA, WlS, WrS, bS, WlE, WrE, bE, Wpack, bpack);

    const int mtiles = nn / 16;   // 3125 (exact)

    // ---- layer 1: x[ nn x64 ] -> hA[ nn x128 ]
    {
        size_t tot = (size_t)ne << 6;
        fill_zero_f32<<<ZB, ZT, 0, stream>>>(msg, (size_t)nn * 64);
        scatter_add<<<(unsigned)((tot + 255) / 256), 256, 0, stream>>>(x, esrc, edst, ne, 6, msg);
        sage_combine_wmma<<<mtiles * (HID / 16), 32, 0, stream>>>(
            msg, rdeg, x, IN_CH, Wl1, Wr1, b1, hA, HID, 0, nn);
    }
    // ---- layer 2: hA -> hB
    {
        size_t tot = (size_t)ne << 7;
        fill_zero_f32<<<ZB, ZT, 0, stream>>>(msg, (size_t)nn * HID);
        scatter_add<<<(unsigned)((tot + 255) / 256), 256, 0, stream>>>(hA, esrc, edst, ne, 7, msg);
        sage_combine_wmma<<<mtiles * (HID / 16), 32, 0, stream>>>(
            msg, rdeg, hA, HID, Wl2, Wr2, b2, hB, HID, 0, nn);
    }
    // ---- layer 3: hB -> hA
    {
        size_t tot = (size_t)ne << 7;
        fill_zero_f32<<<ZB, ZT, 0, stream>>>(msg, (size_t)nn * HID);
        scatter_add<<<(unsigned)((tot + 255) / 256), 256, 0, stream>>>(hB, esrc, edst, ne, 7, msg);
        sage_combine_wmma<<<mtiles * (HID / 16), 32, 0, stream>>>(
            msg, rdeg, hB, HID, Wl3, Wr3, b3, hA, HID, 0, nn);
    }
    // ---- heads: one shared aggregation of hA, one packed 32-col GEMM
    {
        size_t tot = (size_t)ne << 7;
        fill_zero_f32<<<ZB, ZT, 0, stream>>>(msg, (size_t)nn * HID);
        scatter_add<<<(unsigned)((tot + 255) / 256), 256, 0, stream>>>(hA, esrc, edst, ne, 7, msg);
        sage_combine_wmma<<<mtiles * (32 / 16), 32, 0, stream>>>(
            msg, rdeg, hA, HID, Wpack, Wpack + 128 * 32, bpack, out, 32, 1, nn);
    }
}